// SiameseNetworkFewShotClassifier_71236327571543
// MI455X (gfx1250) — compile-verified
//
#include <hip/hip_runtime.h>
#include <hip/hip_bf16.h>
#include <stdint.h>

#define TQ 32        // queries per workgroup
#define TS 128       // supports per workgroup
#define DC 32        // dims per chunk
#define PITCH 36     // floats per LDS row (DC + 4 pad -> 144B, 16B aligned, conflict-free)
#define NTHREADS 256

// Low 32 bits of a generic pointer into LDS are the LDS byte offset on amdgcn
// (addrspacecast AS3->AS0 puts the aperture in the high half).
__device__ __forceinline__ unsigned lds_off(const void* p) {
    return (unsigned)(unsigned long long)p;
}

// CDNA5 async copy: global -> LDS, 16B per lane, GVS addressing
// (uniform 64-bit SGPR base + per-lane 32-bit byte offset).
__device__ __forceinline__ void async_b128(unsigned loff, unsigned voff, const void* base) {
    asm volatile("global_load_async_to_lds_b128 %0, %1, %2 offset:0"
                 :: "v"(loff), "v"(voff), "s"(base)
                 : "memory");
}

template <int N>
__device__ __forceinline__ void wait_async() {
    asm volatile("s_wait_asynccnt %0" :: "i"(N) : "memory");
}

__global__ __launch_bounds__(NTHREADS)
void l1_argmin_kernel(const float* __restrict__ support,
                      const float* __restrict__ query,
                      unsigned long long* __restrict__ gmin,
                      int D) {
    __shared__ __align__(16) float sS[2][TS * PITCH];
    __shared__ __align__(16) float sQ[2][TQ * PITCH];
    __shared__ unsigned long long sMin[TQ];

    const int t  = threadIdx.x;
    const int qg = t & 7;    // 8 groups of 4 queries
    const int sg = t >> 3;   // 32 groups of 4 supports
    const int q_base = blockIdx.x * TQ;
    const int s_base = blockIdx.y * TS;

    if (t < TQ) sMin[t] = ~0ull;

    const unsigned sSOff[2] = { lds_off(&sS[0][0]), lds_off(&sS[1][0]) };
    const unsigned sQOff[2] = { lds_off(&sQ[0][0]), lds_off(&sQ[1][0]) };
    const unsigned rowBytesG = (unsigned)(D * 4);

    const int nchunks = D / DC;

    auto issue = [&](int c, int buf) {
        const unsigned dco = (unsigned)(c * DC * 4);
        // support tile slice: TS*DC floats = 16KB = 1024 x 16B -> 4 per thread
#pragma unroll
        for (int k = 0; k < 4; ++k) {
            unsigned tau = (unsigned)t + (unsigned)(NTHREADS * k);
            unsigned row = tau >> 3, seg = tau & 7u;
            async_b128(sSOff[buf] + row * (PITCH * 4) + seg * 16,
                       (unsigned)(s_base + (int)row) * rowBytesG + dco + seg * 16,
                       support);
        }
        // query tile slice: TQ*DC floats = 4KB = 256 x 16B -> 1 per thread
        {
            unsigned row = (unsigned)t >> 3, seg = (unsigned)t & 7u;
            async_b128(sQOff[buf] + row * (PITCH * 4) + seg * 16,
                       (unsigned)(q_base + (int)row) * rowBytesG + dco + seg * 16,
                       query);
        }
    };

    float acc[4][4];
#pragma unroll
    for (int i = 0; i < 4; ++i)
#pragma unroll
        for (int j = 0; j < 4; ++j) acc[i][j] = 0.0f;

    issue(0, 0);

    for (int c = 0; c < nchunks; ++c) {
        const int cur = c & 1;
        const bool more = (c + 1) < nchunks;
        if (more) issue(c + 1, cur ^ 1);   // prefetch next chunk into other buffer
        if (more) wait_async<5>();         // in-order: current chunk's 5 xfers done
        else      wait_async<0>();
        __syncthreads();                   // all waves' slices landed in LDS

        const float* Sb = &sS[cur][0];
        const float* Qb = &sQ[cur][0];
#pragma unroll
        for (int d4 = 0; d4 < DC / 4; ++d4) {
            float4 q[4], s[4];
#pragma unroll
            for (int i = 0; i < 4; ++i)
                q[i] = *(const float4*)(Qb + (qg * 4 + i) * PITCH + d4 * 4);
#pragma unroll
            for (int j = 0; j < 4; ++j)
                s[j] = *(const float4*)(Sb + (sg * 4 + j) * PITCH + d4 * 4);
#pragma unroll
            for (int i = 0; i < 4; ++i)
#pragma unroll
                for (int j = 0; j < 4; ++j)
                    acc[i][j] += fabsf(q[i].x - s[j].x) + fabsf(q[i].y - s[j].y)
                               + fabsf(q[i].z - s[j].z) + fabsf(q[i].w - s[j].w);
        }
        __syncthreads();                   // everyone done reading `cur` before overwrite
    }

    // Pack (dist_bits << 32) | support_idx: dist >= 0 so f32 bit pattern is
    // order-preserving; low-bits index => lowest index wins ties (matches argmin).
#pragma unroll
    for (int i = 0; i < 4; ++i) {
        unsigned long long best = ~0ull;
#pragma unroll
        for (int j = 0; j < 4; ++j) {
            unsigned sidx = (unsigned)(s_base + sg * 4 + j);
            unsigned long long pk =
                ((unsigned long long)__float_as_uint(acc[i][j]) << 32) | sidx;
            best = (pk < best) ? pk : best;
        }
        atomicMin(&sMin[qg * 4 + i], best);   // ds_min_u64
    }
    __syncthreads();
    if (t < TQ) atomicMin(&gmin[q_base + t], sMin[t]);   // global_atomic_min_u64
}

__global__ void onehot_finalize_kernel(const unsigned long long* __restrict__ gmin,
                                       const int* __restrict__ labels,
                                       float* __restrict__ out,
                                       int NC, int total) {
    int tid = blockIdx.x * blockDim.x + threadIdx.x;
    if (tid >= total) return;
    int q = tid / NC;
    int c = tid - q * NC;
    unsigned idx = (unsigned)gmin[q];        // low 32 bits = argmin support index
    int lab = labels[idx];
    out[tid] = (c == lab) ? 1.0f : 0.0f;
}

extern "C" void kernel_launch(void* const* d_in, const int* in_sizes, int n_in,
                              void* d_out, int out_size, void* d_ws, size_t ws_size,
                              hipStream_t stream) {
    const float* support = (const float*)d_in[0];
    const int*   labels  = (const int*)d_in[1];
    const float* query   = (const float*)d_in[2];

    const int S  = in_sizes[1];            // 1024
    const int D  = in_sizes[0] / S;        // 256
    const int Q  = in_sizes[2] / D;        // 2048
    const int NC = out_size / Q;           // 32

    unsigned long long* gmin = (unsigned long long*)d_ws;
    hipMemsetAsync(gmin, 0xFF, (size_t)Q * sizeof(unsigned long long), stream);

    dim3 grid(Q / TQ, S / TS);             // 64 x 8 = 512 workgroups
    l1_argmin_kernel<<<grid, NTHREADS, 0, stream>>>(support, query, gmin, D);

    const int total = Q * NC;
    onehot_finalize_kernel<<<(total + 255) / 256, 256, 0, stream>>>(
        gmin, labels, (float*)d_out, NC, total);
}